// ComplexUpSampling2D_1967095022301
// MI455X (gfx1250) — compile-verified
//
#include <hip/hip_runtime.h>

// Complex nearest-neighbor 2x2 upsample, channels_last.
// in : x_re[16,128,128,64] f32, x_im[16,128,128,64] f32
// out: complex64 [16,256,256,64] stored as interleaved {re,im} float pairs.
//
// Pure bandwidth problem: 134 MB read + 537 MB written, 0 FLOPs.
// Strategy: read-once / write-4x, 128-bit non-temporal accesses only.

typedef __attribute__((ext_vector_type(4))) float f4;

__global__ __launch_bounds__(256) void complex_upsample2x2_kernel(
    const float* __restrict__ re,
    const float* __restrict__ im,
    float* __restrict__ out)
{
    constexpr unsigned B  = 16, H = 128, W = 128, C = 64;
    constexpr unsigned Ho = H * 2, Wo = W * 2;
    constexpr unsigned CG = C / 4;          // 16 float4 channel-groups per pixel
    (void)B;

    const unsigned tid = blockIdx.x * blockDim.x + threadIdx.x;
    const unsigned cg  = tid & (CG - 1);    // which 4-channel group
    const unsigned pix = tid >> 4;          // flat input pixel (b,ih,iw)
    const unsigned iw  = pix & (W - 1);
    const unsigned ih  = (pix >> 7) & (H - 1);
    const unsigned b   = pix >> 14;

    // ---- load 4 channels of re and im (each exactly once -> NT load) ----
    const unsigned long long in_off =
        (unsigned long long)pix * C + (unsigned long long)cg * 4;
    const f4 r = __builtin_nontemporal_load((const f4*)(re + in_off));
    const f4 i = __builtin_nontemporal_load((const f4*)(im + in_off));

    // ---- interleave into complex pairs: two float4 = 4 complex values ----
    const f4 lo = { r.x, i.x, r.y, i.y };
    const f4 hi = { r.z, i.z, r.w, i.w };

    // ---- output addressing in float4 units (1 float4 == 2 complex) ----
    // complex index of pixel (b,ho,wo), channel 4*cg:
    //   ((b*Ho + ho)*Wo + wo)*C + 4*cg   -> float4 index = that / 2
    constexpr unsigned long long PIX4 = C / 2;        // float4 per output pixel (32)
    constexpr unsigned long long ROW4 = (unsigned long long)Wo * PIX4; // per row

    f4* o = (f4*)out;
    const unsigned long long row0 =
        ((unsigned long long)(b * Ho + 2u * ih) * Wo + 2u * iw) * PIX4
        + (unsigned long long)cg * 2;
    const unsigned long long row1 = row0 + ROW4;

    // ---- broadcast to the 2x2 output block; streaming NT stores ----
    __builtin_nontemporal_store(lo, o + row0);
    __builtin_nontemporal_store(hi, o + row0 + 1);
    __builtin_nontemporal_store(lo, o + row0 + PIX4);
    __builtin_nontemporal_store(hi, o + row0 + PIX4 + 1);
    __builtin_nontemporal_store(lo, o + row1);
    __builtin_nontemporal_store(hi, o + row1 + 1);
    __builtin_nontemporal_store(lo, o + row1 + PIX4);
    __builtin_nontemporal_store(hi, o + row1 + PIX4 + 1);
}

extern "C" void kernel_launch(void* const* d_in, const int* in_sizes, int n_in,
                              void* d_out, int out_size, void* d_ws, size_t ws_size,
                              hipStream_t stream) {
    (void)in_sizes; (void)n_in; (void)out_size; (void)d_ws; (void)ws_size;

    const float* re = (const float*)d_in[0];
    const float* im = (const float*)d_in[1];
    float* out      = (float*)d_out;

    // 16*128*128 input pixels * 16 channel-groups = 4,194,304 threads
    constexpr unsigned total_threads = 16u * 128u * 128u * 16u;
    constexpr unsigned block = 256;                    // 8 waves (wave32)
    constexpr unsigned grid  = total_threads / block;  // 16384, exact

    complex_upsample2x2_kernel<<<grid, block, 0, stream>>>(re, im, out);
}